// Saliency_Sampler_3788161155491
// MI455X (gfx1250) — compile-verified
//
#include <hip/hip_runtime.h>

#define GS   21
#define PADW 20

typedef __attribute__((ext_vector_type(2))) float v2f;
typedef __attribute__((ext_vector_type(8))) float v8f;

// ---------------------------------------------------------------------------
// Kernel 1: 2x2 average pool  x(4,4,1200,1200) -> x600(4,4,600,600)
// Pure streaming, float4 loads / float2 stores. HBM-bound (~115 MB traffic).
// ---------------------------------------------------------------------------
__global__ void k_pool2(const float* __restrict__ x, float* __restrict__ x600) {
  int idx = blockIdx.x * blockDim.x + threadIdx.x;   // 2,880,000 threads, 2 outputs each
  if (idx >= 4 * 4 * 600 * 300) return;
  int xp = idx % 300;
  int y  = (idx / 300) % 600;
  int bc = idx / (300 * 600);
  const float* row0 = x + (size_t)bc * 1440000 + (size_t)(2 * y) * 1200 + 4 * xp;
  float4 r0 = *reinterpret_cast<const float4*>(row0);
  float4 r1 = *reinterpret_cast<const float4*>(row0 + 1200);
  float2 o;
  o.x = 0.25f * (r0.x + r0.y + r1.x + r1.y);
  o.y = 0.25f * (r0.z + r0.w + r1.z + r1.w);
  *reinterpret_cast<float2*>(x600 + (size_t)bc * 360000 + (size_t)y * 600 + 2 * xp) = o;
}

// ---------------------------------------------------------------------------
// Kernel 2: per (batch, branch):
//   xs = resize(pool2(tri),21,21) -> padded 61x61 (stored zero-padded 72x64 LDS)
//   conv41x41 VALID of {xs, xs*basex, xs*basey} via implicit GEMM on
//   v_wmma_f32_16x16x4_f32:  OUT(32x32) = A(32 x 41*64) * B(41*64 x 32)
//     A[oy][fy*64+j] = img_c[oy+fy][j]          (LDS image, zero padded)
//     B[fy*64+j][ox] = w[fy][j-ox]  (Toeplitz; zero-padded filter rows in LDS
//                                    make this an unconditional ds_load)
//   then g = clip(conv_x/conv_s*2-1, -1, 1) -> g21[branch][b][2][21][21]
// 12 waves: channel(3) x Mtile(2) x Ntile(2). 656 chained WMMAs per wave.
// ---------------------------------------------------------------------------
__global__ __launch_bounds__(384) void k_makegrid(const float* __restrict__ tri,
                                                  const float* __restrict__ tri_rev,
                                                  const float* __restrict__ filt,
                                                  float* __restrict__ g21) {
  __shared__ float sImg[3][72 * 64];
  __shared__ float sW[41 * 96];          // row r: [0,32)=0, [32,73)=w[r][*], [73,96)=0
  __shared__ float sRes[3][32 * 32];

  const int tid = threadIdx.x;
  const int branch = blockIdx.x >> 2;
  const int b = blockIdx.x & 3;
  const float* src = (branch == 0 ? tri : tri_rev) + (size_t)b * 1440000;

  for (int i = tid; i < 3 * 72 * 64; i += 384) (&sImg[0][0])[i] = 0.0f;
  for (int i = tid; i < 41 * 96; i += 384) sW[i] = 0.0f;
  __syncthreads();

  // fill filter taps (zero padding already in place)
  for (int i = tid; i < 41 * 41; i += 384) {
    int fy = i / 41, fx = i % 41;
    sW[fy * 96 + 32 + fx] = filt[i];
  }

  if (tid < 441) {
    int i = tid / 21, j = tid % 21;
    // jax half-pixel bilinear 600->21 (interior: no edge handling needed)
    float sy = (i + 0.5f) * (600.0f / 21.0f) - 0.5f;
    float sx = (j + 0.5f) * (600.0f / 21.0f) - 0.5f;
    int y0 = (int)floorf(sy), x0 = (int)floorf(sx);
    float fy = sy - (float)y0, fx = sx - (float)x0;
    float p[2][2];
#pragma unroll
    for (int dy = 0; dy < 2; ++dy)
#pragma unroll
      for (int dx = 0; dx < 2; ++dx) {
        const float* r0 = src + (size_t)(2 * (y0 + dy)) * 1200 + 2 * (x0 + dx);
        const float* r1 = r0 + 1200;
        p[dy][dx] = 0.25f * (r0[0] + r0[1] + r1[0] + r1[1]);  // on-the-fly pool2
      }
    float v = (1.f - fy) * ((1.f - fx) * p[0][0] + fx * p[0][1]) +
              fy * ((1.f - fx) * p[1][0] + fx * p[1][1]);
    int rr = i + PADW, cc = j + PADW;
    float bx = (float)(cc - PADW) * (1.0f / (GS - 1));
    float by = (float)(rr - PADW) * (1.0f / (GS - 1));
    sImg[0][rr * 64 + cc] = v;
    sImg[1][rr * 64 + cc] = v * bx;
    sImg[2][rr * 64 + cc] = v * by;
  }
  __syncthreads();

  {
    const int lane = tid & 31;
    const int l15 = lane & 15;
    const bool hiL = lane >= 16;
    const int wave = tid >> 5;          // 0..11
    const int ch = wave >> 2;           // 0..2
    const int mT = (wave >> 1) & 1;
    const int nT = wave & 1;
    const float* img = sImg[ch];
    const int m = mT * 16 + l15;        // A row (lanes 0-15 carry M)
    const int n = nT * 16 + l15;        // B col
    const int koff = hiL ? 2 : 0;       // lanes 16-31 carry K+2 (ISA A/B f32 layout)
    v8f acc = {0.f, 0.f, 0.f, 0.f, 0.f, 0.f, 0.f, 0.f};
    for (int fyI = 0; fyI < 41; ++fyI) {
      const float* arow  = img + (m + fyI) * 64 + koff;
      const float* wrowP = sW + fyI * 96 + 32 - n + koff;   // padded row, no bounds check
#pragma unroll 4
      for (int jb = 0; jb < 64; jb += 4) {
        v2f a;
        a.x = arow[jb];
        a.y = arow[jb + 1];
        v2f bv;
        bv.x = wrowP[jb];
        bv.y = wrowP[jb + 1];
        acc = __builtin_amdgcn_wmma_f32_16x16x4_f32(false, a, false, bv,
                                                    (short)0, acc, false, false);
      }
    }
#pragma unroll
    for (int t = 0; t < 8; ++t) {   // D layout: VGPR t -> M=t (lo lanes) / t+8 (hi lanes)
      int Ml = t + (hiL ? 8 : 0);
      sRes[ch][(mT * 16 + Ml) * 32 + nT * 16 + l15] = acc[t];
    }
  }
  __syncthreads();

  if (tid < 441) {
    int i = tid / 21, j = tid % 21;
    float s = sRes[0][i * 32 + j];
    float gx = fminf(fmaxf(sRes[1][i * 32 + j] / s * 2.0f - 1.0f, -1.0f), 1.0f);
    float gy = fminf(fmaxf(sRes[2][i * 32 + j] / s * 2.0f - 1.0f, -1.0f), 1.0f);
    size_t base = (size_t)(branch * 4 + b) * 2 * 441;
    g21[base + tid] = gx;
    g21[base + 441 + tid] = gy;
  }
}

// ---------------------------------------------------------------------------
// Kernel 3: fused grid-upsample (21->512, jax half-pixel bilinear; edge
// renormalization == coordinate clamp) + grid_sample of x600 (4 channels,
// zeros padding). mask output == channel-3 sample (mask600 = x600[:,3:4]).
// ---------------------------------------------------------------------------
__global__ void k_sample(const float* __restrict__ x600, const float* __restrict__ g21,
                         float* __restrict__ out) {
  int idx = blockIdx.x * blockDim.x + threadIdx.x;
  if (idx >= 2 * 4 * 512 * 512) return;
  int ox = idx & 511;
  int oy = (idx >> 9) & 511;
  int b  = (idx >> 18) & 3;
  int branch = idx >> 20;

  float sfx = (ox + 0.5f) * (21.0f / 512.0f) - 0.5f;
  float sfy = (oy + 0.5f) * (21.0f / 512.0f) - 0.5f;
  sfx = fminf(fmaxf(sfx, 0.0f), 20.0f);
  sfy = fminf(fmaxf(sfy, 0.0f), 20.0f);
  int jx0 = (int)sfx, jy0 = (int)sfy;   // sf >= 0: trunc == floor
  float tx = sfx - (float)jx0, ty = sfy - (float)jy0;
  int jx1 = min(jx0 + 1, 20), jy1 = min(jy0 + 1, 20);
  const float* g0 = g21 + (size_t)(branch * 4 + b) * 2 * 441;
  const float* g1 = g0 + 441;
  float w00 = (1.f - ty) * (1.f - tx), w01 = (1.f - ty) * tx;
  float w10 = ty * (1.f - tx), w11 = ty * tx;
  float gx = w00 * g0[jy0 * 21 + jx0] + w01 * g0[jy0 * 21 + jx1] +
             w10 * g0[jy1 * 21 + jx0] + w11 * g0[jy1 * 21 + jx1];
  float gy = w00 * g1[jy0 * 21 + jx0] + w01 * g1[jy0 * 21 + jx1] +
             w10 * g1[jy1 * 21 + jx0] + w11 * g1[jy1 * 21 + jx1];

  float ix = ((gx + 1.0f) * 600.0f - 1.0f) * 0.5f;
  float iy = ((gy + 1.0f) * 600.0f - 1.0f) * 0.5f;
  float fx0 = floorf(ix), fy0 = floorf(iy);
  int x0 = (int)fx0, y0 = (int)fy0;
  float wx1 = ix - fx0, wy1 = iy - fy0;
  float wx0 = 1.f - wx1, wy0 = 1.f - wy1;

  float acc0 = 0.f, acc1 = 0.f, acc2 = 0.f, acc3 = 0.f;
  const float* base = x600 + (size_t)b * 4 * 360000;
#pragma unroll
  for (int dy = 0; dy < 2; ++dy) {
    int yc = y0 + dy;
    bool vy = (yc >= 0) && (yc <= 599);
    float wy = dy ? wy1 : wy0;
    int yi = min(max(yc, 0), 599);
#pragma unroll
    for (int dx = 0; dx < 2; ++dx) {
      int xc = x0 + dx;
      bool vx = (xc >= 0) && (xc <= 599);
      float w = wy * (dx ? wx1 : wx0) * ((vx && vy) ? 1.0f : 0.0f);
      int xi = min(max(xc, 0), 599);
      size_t off = (size_t)yi * 600 + xi;
      acc0 += base[0 * 360000 + off] * w;
      acc1 += base[1 * 360000 + off] * w;
      acc2 += base[2 * 360000 + off] * w;
      acc3 += base[3 * 360000 + off] * w;
    }
  }
  size_t ob  = (size_t)branch * 4194304;             // x_sampled / x_sampled1
  size_t pix = (size_t)oy * 512 + ox;
  size_t bb  = (size_t)b * 4 * 262144;
  out[ob + bb + 0 * 262144 + pix] = acc0;
  out[ob + bb + 1 * 262144 + pix] = acc1;
  out[ob + bb + 2 * 262144 + pix] = acc2;
  out[ob + bb + 3 * 262144 + pix] = acc3;
  // mask_sample / mask_sample1 == channel-3 sample
  out[8388608 + (size_t)branch * 1048576 + (size_t)b * 262144 + pix] = acc3;
}

// ---------------------------------------------------------------------------
extern "C" void kernel_launch(void* const* d_in, const int* in_sizes, int n_in,
                              void* d_out, int out_size, void* d_ws, size_t ws_size,
                              hipStream_t stream) {
  const float* x      = (const float*)d_in[0];
  const float* tri    = (const float*)d_in[1];
  const float* trirev = (const float*)d_in[2];
  const float* filt   = (const float*)d_in[3];
  float* out  = (float*)d_out;
  float* x600 = (float*)d_ws;                 // 5,760,000 floats (23.04 MB)
  float* g21  = x600 + 5760000;               // 7,056 floats

  k_pool2<<<(2880000 + 255) / 256, 256, 0, stream>>>(x, x600);
  k_makegrid<<<8, 384, 0, stream>>>(tri, trirev, filt, g21);
  k_sample<<<(2097152 + 255) / 256, 256, 0, stream>>>(x600, g21, out);
}